// GATModel_81106162418331
// MI455X (gfx1250) — compile-verified
//
#include <hip/hip_runtime.h>

// GAT 2-layer forward for MI455X (gfx1250).
// GEMMs use V_WMMA_F32_16X16X4_F32 (exact fp32, matrix pipe).
// Edge softmax/aggregation via atomics; all buffers re-initialized per call.

typedef __attribute__((ext_vector_type(2))) float v2f;
typedef __attribute__((ext_vector_type(8))) float v8f;

#define NEG_SLOPE 0.2f

__device__ __forceinline__ float atomic_max_f32(float* addr, float val) {
  // Standard monotone-encoding trick; valid with -inf init and mixed signs.
  if (val >= 0.0f)
    return __int_as_float(atomicMax((int*)addr, __float_as_int(val)));
  else
    return __uint_as_float(atomicMin((unsigned int*)addr, __float_as_uint(val)));
}

__device__ __forceinline__ void edge_endpoints(int e, int E, const int* __restrict__ ei,
                                               int& s, int& d) {
  if (e < E) { s = ei[e]; d = ei[E + e]; }
  else       { s = e - E; d = s; }          // implicit self-loop
}

// ---------------------------------------------------------------------------
// C[M,N] = A[M,K] @ B[K,N], fp32, one 16x16 tile per wave32.
// M % 16 == 0, N % 16 == 0, K % 4 == 0 (all exact for this model).
// ---------------------------------------------------------------------------
__global__ void gemm_wmma_f32(const float* __restrict__ A, const float* __restrict__ B,
                              float* __restrict__ C, int M, int N, int K) {
  const int lane = threadIdx.x & 31;
  const int wave = threadIdx.x >> 5;
  const int tile = blockIdx.x * (blockDim.x >> 5) + wave;
  const int tiles_n = N >> 4;
  const int tiles   = (M >> 4) * tiles_n;
  if (tile >= tiles) return;                 // uniform per-wave, EXEC stays all-1s

  const int tm = (tile / tiles_n) << 4;
  const int tn = (tile % tiles_n) << 4;
  const int l15 = lane & 15;
  const int kh  = (lane >> 4) << 1;          // 0 for lanes 0-15, 2 for lanes 16-31

  // A frag: lane covers row (tm+l15), K slots kh, kh+1 of each k0 group.
  const float* arow = A + (long)(tm + l15) * K + kh;
  // B frag: lane covers col (tn+l15), K rows kh, kh+1 of each k0 group.
  const float* bcol = B + (long)kh * N + tn + l15;

  v8f acc = {};
  for (int k0 = 0; k0 < K; k0 += 4) {
    v2f a, b;
    a.x = arow[0];       a.y = arow[1];      // consecutive -> b64 load
    b.x = bcol[0];       b.y = bcol[N];
    acc = __builtin_amdgcn_wmma_f32_16x16x4_f32(
        /*neg_a=*/false, a, /*neg_b=*/false, b,
        /*c_mod=*/(short)0, acc, /*reuse_a=*/false, /*reuse_b=*/false);
    arow += 4;
    bcol += 4 * (long)N;
  }

  // D layout: VGPR r -> row r (lanes 0-15) / row r+8 (lanes 16-31), col = lane&15.
  const int row0 = tm + ((lane >> 4) << 3);
  const int col  = tn + l15;
  #pragma unroll
  for (int r = 0; r < 8; ++r)
    C[(long)(row0 + r) * N + col] = acc[r];
}

// ---------------------------------------------------------------------------
// Per (node, head): attention dot products + init of segment max / denom.
// ---------------------------------------------------------------------------
__global__ void node_alpha(const float* __restrict__ h, const float* __restrict__ a_src,
                           const float* __restrict__ a_dst, float* __restrict__ alpha_s,
                           float* __restrict__ alpha_d, float* __restrict__ mmax,
                           float* __restrict__ denom, int N, int H, int D) {
  const int i = blockIdx.x * blockDim.x + threadIdx.x;   // over N*H
  if (i >= N * H) return;
  const int n = i / H, hh = i % H;
  const float* hv = h + (long)n * H * D + (long)hh * D;
  const float* as = a_src + hh * D;
  const float* ad = a_dst + hh * D;
  float ss = 0.0f, sd = 0.0f;
  for (int d = 0; d < D; ++d) { const float v = hv[d]; ss += v * as[d]; sd += v * ad[d]; }
  alpha_s[i] = ss;
  alpha_d[i] = sd;
  mmax[i]  = -__builtin_inff();
  denom[i] = 0.0f;
}

__global__ void fill_f32(float* __restrict__ p, float v, long n) {
  const long i = (long)blockIdx.x * blockDim.x + threadIdx.x;
  if (i < n) p[i] = v;
}

// ---------------------------------------------------------------------------
// Edge pass 1: segment max of leaky-relu logits.
// ---------------------------------------------------------------------------
__global__ void edge_max(const int* __restrict__ ei, int E, int N, int H,
                         const float* __restrict__ alpha_s, const float* __restrict__ alpha_d,
                         float* __restrict__ mmax) {
  const long t = (long)blockIdx.x * blockDim.x + threadIdx.x;
  const long total = (long)(E + N) * H;
  if (t >= total) return;
  const int e = (int)(t / H), hh = (int)(t % H);
  int s, d; edge_endpoints(e, E, ei, s, d);
  float v = alpha_s[s * H + hh] + alpha_d[d * H + hh];
  v = (v > 0.0f) ? v : NEG_SLOPE * v;
  atomic_max_f32(&mmax[d * H + hh], v);
}

// ---------------------------------------------------------------------------
// Edge pass 2: exp(e - max), store per-edge weight, accumulate denom.
// ---------------------------------------------------------------------------
__global__ void edge_exp(const int* __restrict__ ei, int E, int N, int H,
                         const float* __restrict__ alpha_s, const float* __restrict__ alpha_d,
                         const float* __restrict__ mmax, float* __restrict__ denom,
                         float* __restrict__ eexp) {
  const long t = (long)blockIdx.x * blockDim.x + threadIdx.x;
  const long total = (long)(E + N) * H;
  if (t >= total) return;
  const int e = (int)(t / H), hh = (int)(t % H);
  int s, d; edge_endpoints(e, E, ei, s, d);
  float v = alpha_s[s * H + hh] + alpha_d[d * H + hh];
  v = (v > 0.0f) ? v : NEG_SLOPE * v;
  const float w = __expf(v - mmax[d * H + hh]);
  eexp[t] = w;
  atomicAdd(&denom[d * H + hh], w);
}

// ---------------------------------------------------------------------------
// Edge pass 3: one block per edge, blockDim == H*D. Weighted scatter-add.
// ---------------------------------------------------------------------------
__global__ void edge_aggr(const int* __restrict__ ei, int E, int N, int H, int dshift,
                          const float* __restrict__ hfeat, const float* __restrict__ eexp,
                          const float* __restrict__ denom, float* __restrict__ out) {
  __shared__ float s_alpha[4];
  const int e = blockIdx.x;
  int s, d; edge_endpoints(e, E, ei, s, d);
  const int c = threadIdx.x;                       // 0 .. H*D-1
  if (c < H)
    s_alpha[c] = eexp[(long)e * H + c] / (denom[d * H + c] + 1e-16f);
  __syncthreads();
  const int hd = blockDim.x;                       // H*D
  const float val = hfeat[(long)s * hd + c] * s_alpha[c >> dshift];
  atomicAdd(&out[(long)d * hd + c], val);
}

__global__ void bias_act(float* __restrict__ x, const float* __restrict__ b,
                         long n, int cmask, int do_relu) {
  const long i = (long)blockIdx.x * blockDim.x + threadIdx.x;
  if (i >= n) return;
  float v = x[i] + b[i & cmask];
  if (do_relu) v = (v > 0.0f) ? v : 0.0f;
  x[i] = v;
}

// ---------------------------------------------------------------------------
// Launch
// ---------------------------------------------------------------------------
extern "C" void kernel_launch(void* const* d_in, const int* in_sizes, int n_in,
                              void* d_out, int out_size, void* d_ws, size_t ws_size,
                              hipStream_t stream) {
  const float* x   = (const float*)d_in[0];
  const int*   ei  = (const int*)  d_in[1];
  const float* W1  = (const float*)d_in[2];
  const float* as1 = (const float*)d_in[3];
  const float* ad1 = (const float*)d_in[4];
  const float* b1  = (const float*)d_in[5];
  const float* W2  = (const float*)d_in[6];
  const float* as2 = (const float*)d_in[7];
  const float* ad2 = (const float*)d_in[8];
  const float* b2  = (const float*)d_in[9];
  float* out = (float*)d_out;

  const int IN = 256, H = 4, D = 64, HD = H * D;   // 256
  const int N  = in_sizes[0] / IN;                  // 50000
  const int E  = in_sizes[1] / 2;                   // 800000
  const int NE = E + N;                             // edges incl. self-loops

  // Workspace carve-out (fp32 elements). h2/ee2/layer-2 scalars alias layer-1
  // buffers that are dead by the time they're written.
  float* ws = (float*)d_ws;
  float* h1   = ws;  ws += (size_t)N * HD;          // layer-1 features [N,256]
  float* out1 = ws;  ws += (size_t)N * HD;          // layer-1 output   [N,256]
  float* al_s1 = ws; ws += (size_t)N * H;
  float* al_d1 = ws; ws += (size_t)N * H;
  float* m1    = ws; ws += (size_t)N * H;
  float* dn1   = ws; ws += (size_t)N * H;
  float* ee1   = ws; ws += (size_t)NE * H;          // per-edge exp weights
  float* h2    = h1;                                // [N,64] aliases dead h1
  float* al_s2 = al_s1;                             // [N] aliases layer-1 (dead)
  float* al_d2 = al_d1;
  float* m2    = m1;
  float* dn2   = dn1;
  float* ee2   = ee1;

  const int TB = 256;

  // ---- Layer 1 ----
  {
    const int tiles = (N / 16) * (HD / 16);
    gemm_wmma_f32<<<(tiles + 3) / 4, 128, 0, stream>>>(x, W1, h1, N, HD, IN);
  }
  {
    const long n = (long)N * H;
    node_alpha<<<(int)((n + TB - 1) / TB), TB, 0, stream>>>(h1, as1, ad1, al_s1, al_d1,
                                                            m1, dn1, N, H, D);
  }
  {
    const long n = (long)N * HD;
    fill_f32<<<(int)((n + TB - 1) / TB), TB, 0, stream>>>(out1, 0.0f, n);
  }
  {
    const long n = (long)NE * H;
    edge_max<<<(int)((n + TB - 1) / TB), TB, 0, stream>>>(ei, E, N, H, al_s1, al_d1, m1);
    edge_exp<<<(int)((n + TB - 1) / TB), TB, 0, stream>>>(ei, E, N, H, al_s1, al_d1, m1,
                                                          dn1, ee1);
  }
  edge_aggr<<<NE, HD, 0, stream>>>(ei, E, N, H, /*dshift=*/6, h1, ee1, dn1, out1);
  {
    const long n = (long)N * HD;
    bias_act<<<(int)((n + TB - 1) / TB), TB, 0, stream>>>(out1, b1, n, HD - 1, /*relu=*/1);
  }

  // ---- Layer 2 (H=1) ----
  {
    const int tiles = (N / 16) * (D / 16);
    gemm_wmma_f32<<<(tiles + 3) / 4, 128, 0, stream>>>(out1, W2, h2, N, D, HD);
  }
  {
    const long n = (long)N;
    node_alpha<<<(int)((n + TB - 1) / TB), TB, 0, stream>>>(h2, as2, ad2, al_s2, al_d2,
                                                            m2, dn2, N, 1, D);
  }
  {
    const long n = (long)N * D;
    fill_f32<<<(int)((n + TB - 1) / TB), TB, 0, stream>>>(out, 0.0f, n);
  }
  {
    const long n = (long)NE;
    edge_max<<<(int)((n + TB - 1) / TB), TB, 0, stream>>>(ei, E, N, 1, al_s2, al_d2, m2);
    edge_exp<<<(int)((n + TB - 1) / TB), TB, 0, stream>>>(ei, E, N, 1, al_s2, al_d2, m2,
                                                          dn2, ee2);
  }
  edge_aggr<<<NE, D, 0, stream>>>(ei, E, N, 1, /*dshift=*/6, h2, ee2, dn2, out);
  {
    const long n = (long)N * D;
    bias_act<<<(int)((n + TB - 1) / TB), TB, 0, stream>>>(out, b2, n, D - 1, /*relu=*/0);
  }
}